// Seq2SeqLARS_VSA_43087111913960
// MI455X (gfx1250) — compile-verified
//
#include <hip/hip_runtime.h>
#include <hip/hip_fp16.h>

typedef __attribute__((ext_vector_type(2))) float v2f;
typedef __attribute__((ext_vector_type(8))) float v8f;

constexpr int NB    = 4096;   // batch
constexpr int NS    = 5;      // seq positions
constexpr int ND    = 64;     // signal length (GEMM K)
constexpr int NHS   = 961;    // kernel taps
constexpr int NVSA  = 1024;   // output features (GEMM N)
constexpr int HPADL = 63;     // zero guard each side of h
constexpr int HPITCH= 1088;   // 961 + 126 rounded up
constexpr float EPS = 1e-3f;
constexpr int BB    = 16;     // batches per block (WMMA M)
constexpr int NT    = NVSA / 16;   // 64 n-tiles of 16
constexpr int T2    = 512;    // threads in fused kernel
constexpr int W2    = T2 / 32;

__device__ __forceinline__ float sgnf(float x) {
    return (x > 0.f) ? 1.f : ((x < 0.f) ? -1.f : 0.f);
}
// silu via v_exp_f32 + v_rcp_f32 (TRANS ops, co-execute with VALU); avoids the
// ~10-op IEEE divide sequence the plain x/(1+e) lowers to.
__device__ __forceinline__ float siluf(float x) {
    return x * __builtin_amdgcn_rcpf(1.f + __expf(-x));
}

// ---------------------------------------------------------------------------
// Kernel 1: broadcast branch. sp_ln[j, t] for j<5, t<1024 (conv+silu+LN).
// Tiny (5*1024 outputs); plain VALU is fine.
// ---------------------------------------------------------------------------
__global__ void sp_conv_ln_kernel(const float* __restrict__ h,
                                  const float* __restrict__ symbols,
                                  const float* __restrict__ gamma,
                                  const float* __restrict__ beta,
                                  float* __restrict__ spl) {
    __shared__ float hp[HPITCH];
    __shared__ float sym[ND];
    __shared__ float yv[NVSA];
    __shared__ float red[2];
    __shared__ float mv[2];
    const int j = blockIdx.x;
    const int tid = threadIdx.x;

    for (int k = tid; k < HPITCH; k += blockDim.x) hp[k] = 0.f;
    if (tid < 2) red[tid] = 0.f;
    __syncthreads();
    for (int k = tid; k < NHS; k += blockDim.x) hp[HPADL + k] = h[j * NHS + k];
    for (int k = tid; k < ND; k += blockDim.x) sym[k] = symbols[j * ND + k];
    __syncthreads();

    float lsum = 0.f, lsq = 0.f;
    for (int t = tid; t < NVSA; t += blockDim.x) {
        const float* hh = &hp[1023 - t];     // hp[1023-t+d] == h[960-t+d]
        float acc = 0.f;
        #pragma unroll 8
        for (int d = 0; d < ND; ++d) acc += sym[d] * hh[d];
        float y = siluf(acc);
        yv[t] = y;
        lsum += y; lsq += y * y;
    }
    atomicAdd(&red[0], lsum);
    atomicAdd(&red[1], lsq);
    __syncthreads();
    if (tid == 0) {
        float mu = red[0] * (1.f / NVSA);
        float var = red[1] * (1.f / NVSA) - mu * mu;
        mv[0] = mu;
        mv[1] = rsqrtf(var + EPS);
    }
    __syncthreads();
    const float mu = mv[0], rs = mv[1];
    for (int t = tid; t < NVSA; t += blockDim.x)
        spl[j * NVSA + t] = (yv[t] - mu) * rs * gamma[t] + beta[t];
}

// ---------------------------------------------------------------------------
// Kernel 2: fully fused. One block = 16 batches x all 5 seq x all 1024 feats.
//   Phase 1: Toeplitz GEMM via V_WMMA_F32_16X16X4_F32 (K=64 -> 16 wmma/tile),
//            silu, f32 LN stats via shuffles + ds_add_f32, vp -> LDS f16.
//   Phase 2: LN params.  Phase 2.5: normalize vph in place (LN applied ONCE).
//   Phase 3: sign scores. Phase 4: softmax(5).
//   Phase 5: att = scores x vp, out = silu(att * sp) -> HBM (only write).
// ---------------------------------------------------------------------------
__global__ void __launch_bounds__(T2)
fused_vsa_kernel(const float* __restrict__ values,
                 const float* __restrict__ h,
                 const float* __restrict__ gamma,
                 const float* __restrict__ beta,
                 const float* __restrict__ spl_g,
                 float* __restrict__ out) {
    extern __shared__ __align__(16) char smem_raw[];
    float* hp   = (float*)smem_raw;          // NS*HPITCH  padded h
    float* spl  = hp   + NS * HPITCH;        // NS*NVSA    sp (post-LN)
    float* gam  = spl  + NS * NVSA;          // NVSA
    float* bet  = gam  + NVSA;               // NVSA
    float* Als  = bet  + NVSA;               // BB*NS*ND   A staging [r][i][d]
    float* ssum = Als  + BB * NS * ND;       // BB*NS
    float* ssq  = ssum + BB * NS;            // BB*NS
    float* muv  = ssq  + BB * NS;            // BB*NS
    float* rsv  = muv  + BB * NS;            // BB*NS
    float* scor = rsv  + BB * NS;            // BB*NS*NS
    float* prob = scor + BB * NS * NS;       // BB*NS*NS
    __half* vph = (__half*)(prob + BB * NS * NS);   // BB*NS*NVSA

    const int tid  = threadIdx.x;
    const int lane = tid & 31;
    const int wave = tid >> 5;
    const int b0   = blockIdx.x * BB;
    const int half = lane >> 4;              // 0: lanes 0-15, 1: lanes 16-31
    const int rn   = lane & 15;              // row (A/C) or col (B/C) index

    // ---- Phase 0: stage everything into LDS ----
    for (int k = tid; k < NS * HPITCH; k += T2) hp[k] = 0.f;
    for (int k = tid; k < BB * NS; k += T2) { ssum[k] = 0.f; ssq[k] = 0.f; }
    __syncthreads();
    for (int k = tid; k < NS * NHS; k += T2) {
        int i = k / NHS, kk = k % NHS;
        hp[i * HPITCH + HPADL + kk] = h[k];
    }
    for (int k = tid; k < NS * NVSA; k += T2) spl[k] = spl_g[k];
    for (int k = tid; k < NVSA; k += T2) { gam[k] = gamma[k]; bet[k] = beta[k]; }
    for (int k = tid; k < BB * NS * ND; k += T2)
        Als[k] = values[(size_t)b0 * NS * ND + k];
    __syncthreads();

    // ---- Phase 1: WMMA GEMM + silu + stats ----
    for (int tile = wave; tile < NS * NT; tile += W2) {
        const int i  = tile / NT;
        const int t0 = (tile % NT) * 16;

        // A fragments: row = rn, K split by lane half (ISA 16x4 f32 layout)
        v2f af[16];
        const float* arow = &Als[(rn * NS + i) * ND];
        #pragma unroll
        for (int kk = 0; kk < 16; ++kk) {
            const int ko = kk * 4 + half * 2;
            af[kk].x = arow[ko];
            af[kk].y = arow[ko + 1];
        }
        // B columns are contiguous slices of padded h: B[k][n] = hp_i[1023-t0-n+k]
        const float* hb = &hp[i * HPITCH + (1023 - t0 - rn)];
        v8f c = {};
        #pragma unroll
        for (int kk = 0; kk < 16; ++kk) {
            const int ko = kk * 4 + half * 2;
            v2f bf;
            bf.x = hb[ko];
            bf.y = hb[ko + 1];
            c = __builtin_amdgcn_wmma_f32_16x16x4_f32(
                    false, af[kk], false, bf, (short)0, c, false, false);
        }
        // silu, f16 stash, per-row sum/sumsq (lanes of one half share a row set)
        #pragma unroll
        for (int p = 0; p < 8; ++p) {
            const float y = siluf(c[p]);
            const int r = p + 8 * half;          // batch row within tile
            vph[(r * NS + i) * NVSA + (t0 + rn)] = __float2half(y);
            float s = y, q = y * y;
            #pragma unroll
            for (int m = 1; m < 16; m <<= 1) {
                s += __shfl_xor(s, m, 16);
                q += __shfl_xor(q, m, 16);
            }
            if (rn == 0) {
                atomicAdd(&ssum[r * NS + i], s);
                atomicAdd(&ssq[r * NS + i], q);
            }
        }
    }
    __syncthreads();

    // ---- Phase 2: LN parameters ----
    if (tid < BB * NS) {
        const float mu  = ssum[tid] * (1.f / NVSA);
        const float var = ssq[tid] * (1.f / NVSA) - mu * mu;
        muv[tid] = mu;
        rsv[tid] = rsqrtf(var + EPS);
    }
    __syncthreads();

    // ---- Phase 2.5: apply LayerNorm to vph in place (exactly once) ----
    for (int idx = tid; idx < BB * NS * NVSA; idx += T2) {
        const int t = idx & (NVSA - 1);
        const int r = idx >> 10;                 // b*NS + i
        const float xi =
            (__half2float(vph[idx]) - muv[r]) * rsv[r] * gam[t] + bet[t];
        vph[idx] = __float2half(xi);
    }
    __syncthreads();

    // ---- Phase 3: sign-agreement scores (one wave per (b,j,i) combo) ----
    for (int combo = wave; combo < BB * NS * NS; combo += W2) {
        const int b = combo / (NS * NS);
        const int rem = combo % (NS * NS);
        const int j = rem / NS;
        const int i = rem % NS;
        const __half* vrow = &vph[(b * NS + i) * NVSA];
        const float*  srow = &spl[j * NVSA];
        float acc = 0.f;
        for (int t = lane; t < NVSA; t += 32) {
            const float xi = __half2float(vrow[t]);
            acc += sgnf(xi) * sgnf(xi + srow[t]);
        }
        #pragma unroll
        for (int m = 1; m < 32; m <<= 1) acc += __shfl_xor(acc, m, 32);
        if (lane == 0) scor[combo] = acc * (1.f / NVSA);
    }
    __syncthreads();

    // ---- Phase 4: softmax over i (size 5) ----
    if (tid < BB * NS) {
        const int b = tid / NS, j = tid % NS;
        float sv[NS], mx = -1e30f;
        #pragma unroll
        for (int i = 0; i < NS; ++i) {
            sv[i] = scor[(b * NS + j) * NS + i];
            mx = fmaxf(mx, sv[i]);
        }
        float sum = 0.f;
        #pragma unroll
        for (int i = 0; i < NS; ++i) { sv[i] = __expf(sv[i] - mx); sum += sv[i]; }
        const float inv = __builtin_amdgcn_rcpf(sum);
        #pragma unroll
        for (int i = 0; i < NS; ++i) prob[(b * NS + j) * NS + i] = sv[i] * inv;
    }
    __syncthreads();

    // ---- Phase 5: attention + output (only HBM write) ----
    for (int idx = tid; idx < BB * NS * NVSA; idx += T2) {
        const int t  = idx & (NVSA - 1);
        const int bj = idx >> 10;
        const int b  = bj / NS, j = bj % NS;
        float att = 0.f;
        #pragma unroll
        for (int i = 0; i < NS; ++i)
            att += prob[(b * NS + j) * NS + i] *
                   __half2float(vph[(b * NS + i) * NVSA + t]);
        const float v = att * spl[j * NVSA + t];
        out[((size_t)(b0 + b) * NS + j) * NVSA + t] = siluf(v);
    }
}

extern "C" void kernel_launch(void* const* d_in, const int* in_sizes, int n_in,
                              void* d_out, int out_size, void* d_ws, size_t ws_size,
                              hipStream_t stream) {
    const float* values  = (const float*)d_in[0];
    const float* h       = (const float*)d_in[1];
    const float* symbols = (const float*)d_in[2];
    const float* gamma   = (const float*)d_in[3];
    const float* beta    = (const float*)d_in[4];
    float* out = (float*)d_out;
    float* spl = (float*)d_ws;   // 5*1024 f32 broadcast sp

    sp_conv_ln_kernel<<<NS, 256, 0, stream>>>(h, symbols, gamma, beta, spl);

    const size_t smem =
        (size_t)(NS * HPITCH + NS * NVSA + 2 * NVSA + BB * NS * ND +
                 4 * BB * NS + 2 * BB * NS * NS) * sizeof(float) +
        (size_t)BB * NS * NVSA * sizeof(__half);   // ~234 KB < 320 KB WGP LDS

    fused_vsa_kernel<<<NB / BB, T2, smem, stream>>>(values, h, gamma, beta,
                                                    spl, out);
}